// SAGE_52673478918617
// MI455X (gfx1250) — compile-verified
//
#include <hip/hip_runtime.h>

typedef float v2f __attribute__((ext_vector_type(2)));
typedef float v8f __attribute__((ext_vector_type(8)));

#define SAGE_N_NODES 100000
#define SAGE_N_EDGES 1600000

// Async global->LDS staging (gfx1250 GLOBAL_LOAD_ASYNC_TO_LDS_B128, ASYNCcnt)
#if defined(__has_builtin)
#if __has_builtin(__builtin_amdgcn_global_load_async_to_lds_b128) && \
    __has_builtin(__builtin_amdgcn_s_wait_asynccnt)
#define SAGE_ASYNC_LDS 1
#endif
#endif
#ifndef SAGE_ASYNC_LDS
#define SAGE_ASYNC_LDS 0
#endif

#if SAGE_ASYNC_LDS
typedef int sage_v4i __attribute__((vector_size(16)));
typedef __attribute__((address_space(1))) sage_v4i* sage_ga_p;
typedef __attribute__((address_space(3))) sage_v4i* sage_lds_p;
__device__ __forceinline__ void sage_async_cp16(const float* g, float* l) {
  __builtin_amdgcn_global_load_async_to_lds_b128(
      (sage_ga_p)(void*)g, (sage_lds_p)(void*)l, 0, 0);
}
#endif

// ---------------------------------------------------------------- utilities
__global__ void sage_fill_zero(float* __restrict__ p, int n) {
  int i = blockIdx.x * blockDim.x + threadIdx.x;
  int stride = gridDim.x * blockDim.x;
  for (; i < n; i += stride) p[i] = 0.0f;
}

__global__ void sage_degree_count(const int* __restrict__ dst, float* __restrict__ deg) {
  int i = blockIdx.x * blockDim.x + threadIdx.x;
  if (i < SAGE_N_EDGES) atomicAdd(&deg[dst[i]], 1.0f);
}

__global__ void sage_inv_deg(float* __restrict__ deg) {
  int i = blockIdx.x * blockDim.x + threadIdx.x;
  if (i < SAGE_N_NODES) deg[i] = 1.0f / fmaxf(deg[i], 1.0f);
}

// ------------------------------------------------- edge scatter (mean-agg numerator)
// One float4 per thread; for FO=128 a full wave32 covers one 512B row (coalesced).
template <int FO>
__global__ __launch_bounds__(256) void sage_scatter_add(
    const float* __restrict__ P, const int* __restrict__ src,
    const int* __restrict__ dst, float* __restrict__ agg) {
  constexpr int TPE = FO / 4;  // threads per edge
  int tid = blockIdx.x * blockDim.x + threadIdx.x;
  int e = tid / TPE;
  int c = tid % TPE;
  if (e >= SAGE_N_EDGES) return;
  int s = src[e];
  int d = dst[e];
  const float4 v = reinterpret_cast<const float4*>(P)[(size_t)s * TPE + c];
  float* base = agg + (size_t)d * FO + (size_t)c * 4;
  atomicAdd(base + 0, v.x);
  atomicAdd(base + 1, v.y);
  atomicAdd(base + 2, v.z);
  atomicAdd(base + 3, v.w);
}

// ------------------------------------------------- WMMA f32 GEMM with fused epilogue
// C[M,N] = A[M,K] @ W[K,N]  (+ bias + agg*inv_deg, optional ReLU)
// 256 threads = 8 wave32; 32 rows per block; K staged through LDS in chunks of 64
// via async global->LDS copies. LDS strides chosen conflict-free & 16B aligned:
//   A stride KC+4: bank = (4*ml + k + 2*half) % 64  -> 32 distinct banks
//   W stride N+8 : lower-half lanes banks [b..b+15], upper [b+16..b+31] -> disjoint
template <int K, int N, bool EPI, bool RELU>
__global__ __launch_bounds__(256) void sage_gemm(
    const float* __restrict__ A, const float* __restrict__ W,
    const float* __restrict__ bias, const float* __restrict__ agg,
    const float* __restrict__ inv_deg, float* __restrict__ C) {
  constexpr int RB = 32;              // output rows per block
  constexpr int KC = 64;              // K chunk staged in LDS
  constexpr int NT = N / 16;          // N tiles
  constexpr int TILES = (RB / 16) * NT;
  constexpr int TPW = TILES / 8;      // tiles per wave (2 for N=128, 1 for N=64)
  constexpr int LAS = KC + 4;         // LDS A row stride (floats)
  constexpr int LWS = N + 8;          // LDS W row stride (floats)

  __shared__ float lA[RB * LAS];
  __shared__ float lW[KC * LWS];

  const int t = threadIdx.x;
  const int wave = t >> 5;
  const int lane = t & 31;
  const int half = lane >> 4;   // 0: K even pair, 1: K odd pair (per ISA A layout)
  const int ml = lane & 15;
  const int rowBase = blockIdx.x * RB;

  v8f acc[TPW];
#pragma unroll
  for (int i = 0; i < TPW; ++i) acc[i] = (v8f){};

  for (int kc = 0; kc < K; kc += KC) {
    // stage A[rowBase:rowBase+32, kc:kc+KC] and W[kc:kc+KC, :] into LDS
#if SAGE_ASYNC_LDS
    for (int i = t; i < RB * KC / 4; i += 256) {
      int r = (i * 4) / KC;
      int c = (i * 4) % KC;
      sage_async_cp16(A + (size_t)(rowBase + r) * K + kc + c, &lA[r * LAS + c]);
    }
    for (int i = t; i < KC * N / 4; i += 256) {
      int r = (i * 4) / N;
      int c = (i * 4) % N;
      sage_async_cp16(W + (size_t)(kc + r) * N + c, &lW[r * LWS + c]);
    }
    __builtin_amdgcn_s_wait_asynccnt(0);
#else
    for (int i = t; i < RB * KC / 4; i += 256) {
      int r = (i * 4) / KC;
      int c = (i * 4) % KC;
      float4 v = *reinterpret_cast<const float4*>(A + (size_t)(rowBase + r) * K + kc + c);
      float* dl = &lA[r * LAS + c];
      dl[0] = v.x; dl[1] = v.y; dl[2] = v.z; dl[3] = v.w;
    }
    for (int i = t; i < KC * N / 4; i += 256) {
      int r = (i * 4) / N;
      int c = (i * 4) % N;
      float4 v = *reinterpret_cast<const float4*>(W + (size_t)(kc + r) * N + c);
      float* dl = &lW[r * LWS + c];
      dl[0] = v.x; dl[1] = v.y; dl[2] = v.z; dl[3] = v.w;
    }
#endif
    __syncthreads();

#pragma unroll
    for (int tw = 0; tw < TPW; ++tw) {
      const int tile = tw * 8 + wave;
      const int wm = tile / NT;
      const int wn = tile % NT;
      const float* Arow = &lA[(wm * 16 + ml) * LAS];
      const int colL = wn * 16 + ml;
#pragma unroll
      for (int k = 0; k < KC; k += 4) {
        v2f a, b;
        // A 16x4 f32 layout: VGPR0 = K+0 (lanes 0-15) / K+2 (lanes 16-31); VGPR1 = K+1 / K+3
        a.x = Arow[k + 2 * half];
        a.y = Arow[k + 2 * half + 1];
        // B 4x16 f32: rows striped across lanes, mirrored split across lane halves
        b.x = lW[(k + 2 * half) * LWS + colL];
        b.y = lW[(k + 2 * half + 1) * LWS + colL];
        acc[tw] = __builtin_amdgcn_wmma_f32_16x16x4_f32(
            false, a, false, b, (short)0, acc[tw], false, false);
      }
    }
    __syncthreads();
  }

  // epilogue + store (C/D layout: VGPR r -> M = r + 8*half, lanes -> N)
#pragma unroll
  for (int tw = 0; tw < TPW; ++tw) {
    const int tile = tw * 8 + wave;
    const int wm = tile / NT;
    const int wn = tile % NT;
    const int col = wn * 16 + ml;
#pragma unroll
    for (int r = 0; r < 8; ++r) {
      const int row = rowBase + wm * 16 + r + 8 * half;
      float v = acc[tw][r];
      if (EPI) {
        v += bias[col] + agg[(size_t)row * N + col] * inv_deg[row];
      }
      if (RELU) v = fmaxf(v, 0.0f);
      C[(size_t)row * N + col] = v;
    }
  }
}

// ---------------------------------------------------------------- launch
extern "C" void kernel_launch(void* const* d_in, const int* in_sizes, int n_in,
                              void* d_out, int out_size, void* d_ws, size_t ws_size,
                              hipStream_t stream) {
  (void)in_sizes; (void)n_in; (void)out_size; (void)ws_size;
  const float* x   = (const float*)d_in[0];
  const int*   src = (const int*)d_in[1];
  const int*   dst = (const int*)d_in[2];
  const float* Ws0 = (const float*)d_in[3];
  const float* Wn0 = (const float*)d_in[4];
  const float* b0  = (const float*)d_in[5];
  const float* Ws1 = (const float*)d_in[6];
  const float* Wn1 = (const float*)d_in[7];
  const float* b1  = (const float*)d_in[8];
  const float* Ws2 = (const float*)d_in[9];
  const float* Wn2 = (const float*)d_in[10];
  const float* b2  = (const float*)d_in[11];
  float* out = (float*)d_out;

  float* ws  = (float*)d_ws;
  float* inv = ws;                                   // N_NODES (padded)
  float* H   = ws + 102400;                          // N_NODES*128
  float* P   = H + (size_t)SAGE_N_NODES * 128;       // N_NODES*128
  float* AGG = P + (size_t)SAGE_N_NODES * 128;       // N_NODES*128

  const int gemmBlocks = SAGE_N_NODES / 32;  // 3125 (exact)
  const int edgeB128 = (SAGE_N_EDGES * 32 + 255) / 256;
  const int edgeB64  = (SAGE_N_EDGES * 16 + 255) / 256;

  // degrees -> inv_deg
  sage_fill_zero<<<512, 256, 0, stream>>>(inv, SAGE_N_NODES);
  sage_degree_count<<<(SAGE_N_EDGES + 255) / 256, 256, 0, stream>>>(dst, inv);
  sage_inv_deg<<<(SAGE_N_NODES + 255) / 256, 256, 0, stream>>>(inv);

  // ----- layer 0: project-then-aggregate, fused self+agg+bias+relu
  sage_gemm<128, 128, false, false><<<gemmBlocks, 256, 0, stream>>>(
      x, Wn0, nullptr, nullptr, nullptr, P);
  sage_fill_zero<<<4096, 256, 0, stream>>>(AGG, SAGE_N_NODES * 128);
  sage_scatter_add<128><<<edgeB128, 256, 0, stream>>>(P, src, dst, AGG);
  sage_gemm<128, 128, true, true><<<gemmBlocks, 256, 0, stream>>>(
      x, Ws0, b0, AGG, inv, H);

  // ----- layer 1 (input H; h2 written into P after scatter consumes it)
  sage_gemm<128, 128, false, false><<<gemmBlocks, 256, 0, stream>>>(
      H, Wn1, nullptr, nullptr, nullptr, P);
  sage_fill_zero<<<4096, 256, 0, stream>>>(AGG, SAGE_N_NODES * 128);
  sage_scatter_add<128><<<edgeB128, 256, 0, stream>>>(P, src, dst, AGG);
  sage_gemm<128, 128, true, true><<<gemmBlocks, 256, 0, stream>>>(
      H, Ws1, b1, AGG, inv, P);

  // ----- layer 2 (input P=h2, OUT=64; projection into H, final into d_out)
  sage_gemm<128, 64, false, false><<<gemmBlocks, 256, 0, stream>>>(
      P, Wn2, nullptr, nullptr, nullptr, H);
  sage_fill_zero<<<4096, 256, 0, stream>>>(AGG, SAGE_N_NODES * 64);
  sage_scatter_add<64><<<edgeB64, 256, 0, stream>>>(H, src, dst, AGG);
  sage_gemm<128, 64, true, false><<<gemmBlocks, 256, 0, stream>>>(
      P, Ws2, b2, AGG, inv, out);
}